// V1_GraphODENet_30769145708811
// MI455X (gfx1250) — compile-verified
//
#include <hip/hip_runtime.h>
#include <math.h>

typedef __attribute__((ext_vector_type(2))) float v2f;
typedef __attribute__((ext_vector_type(4))) float v4f;
typedef __attribute__((ext_vector_type(8))) float v8f;

typedef int v4i_vs __attribute__((vector_size(16)));                 // matches builtin pointee
typedef __attribute__((address_space(1))) v4i_vs* as1_v4i;
typedef __attribute__((address_space(3))) v4i_vs* as3_v4i;

#define DIMV 8192
#define SUBMAX 2048              // max K rows of hh staged in LDS at once
#define WCOLS 64                 // columns owned by one wave (4 x 16-col WMMA tiles)

// ---------- CDNA5 async global->LDS helpers --------------------------------
__device__ __forceinline__ void async_load_b128(const float* g, float* l) {
#if __has_builtin(__builtin_amdgcn_global_load_async_to_lds_b128)
  __builtin_amdgcn_global_load_async_to_lds_b128(
      (as1_v4i)(v4i_vs*)g,            // global src (addrspace 1, int4 pointee)
      (as3_v4i)(v4i_vs*)l,            // LDS dst (addrspacecast generic->3)
      0, 0);
#else
  unsigned loff = (unsigned)(uintptr_t)(__attribute__((address_space(3))) void*)(void*)l;
  asm volatile("global_load_async_to_lds_b128 %0, %1, off"
               :: "v"(loff), "v"(g) : "memory");
#endif
}

template <int N>
__device__ __forceinline__ void wait_async() {
#if __has_builtin(__builtin_amdgcn_s_wait_asynccnt)
  __builtin_amdgcn_s_wait_asynccnt(N);
#else
  asm volatile("s_wait_asynccnt %0" :: "n"(N) : "memory");
#endif
}

// Queue one TKE x 64 W-tile into an LDS buffer: TKE/2 async b128 ops,
// each moving 2 rows (lanes 0-15 -> row 2i, lanes 16-31 -> row 2i+1).
template <int TKE>
__device__ __forceinline__ void issue_wbuf(const float* Wg, float* buf,
                                           int rh, int nl) {
  const float* g = Wg + (size_t)rh * DIMV + 4 * nl;
  float* l = buf + rh * WCOLS + 4 * nl;
#pragma unroll
  for (int i = 0; i < TKE / 2; ++i)
    async_load_b128(g + (size_t)(2 * i) * DIMV, l + 2 * i * WCOLS);
}

// out = tanh( (M @ h) @ W + b ),  M = D^-1/2 (A+I) D^-1/2 of the fixed club graph.
// Wave = 64 cols as 4 interleaved 16-col WMMA tiles; K split NSPLIT ways.
// W streams HBM -> LDS via async b128 (ASYNCcnt ping-pong), LDS -> VGPR via
// conflict-free ds_load_b128; fp32 partials to ws; epilogue sums + bias + tanh.
template <int NSPLIT>
__global__ __launch_bounds__(256) void gcn_partial_wmma(
    const float* __restrict__ h,
    const float* __restrict__ W,
    float* __restrict__ dst)
{
  constexpr int LEN   = DIMV / NSPLIT;                 // K rows per wave
  constexpr int SUB   = (LEN < SUBMAX) ? LEN : SUBMAX; // hh staging chunk
  constexpr int PITCH = SUB + 8;                       // hh bank stagger
  constexpr int TKE   = (NSPLIT >= 16) ? 32 : 16;      // rows per W buffer
  constexpr int NB    = SUB / TKE;                     // buffers per chunk

  __shared__ float hh[4 * PITCH];
  __shared__ __align__(16) float wbuf[8 * 2 * TKE * WCOLS];  // per-wave double buffers

  const int tid  = threadIdx.x;
  const int lane = tid & 31;
  const int wave = tid >> 5;
  const int blk  = blockIdx.x;
  const int ks   = blk % NSPLIT;                 // K slice
  const int grp  = (blk / NSPLIT) * 8 + wave;    // 0..127 column group
  const int j    = grp * WCOLS;

  // Normalized adjacency rows (deg = 4,2,3,3):
  const float M00 = 0.25f;
  const float M01 = 0.35355339059327373f;        // 1/(2*sqrt(2))
  const float M02 = 0.28867513459481287f;        // 1/(2*sqrt(3))
  const float Mt  = 0.33333333333333333f;        // 1/3

  v8f c0 = {}, c1 = {}, c2 = {}, c3 = {};

  const int nl    = lane & 15;                   // column selector
  const int rh    = lane >> 4;                   // row-half (0/1)
  const int khalf = rh << 1;                     // +0 / +2 K offset
  const int arow  = lane & 3;                    // A rows >=4 are don't-care

  float* buf0 = &wbuf[(wave * 2 + 0) * TKE * WCOLS];
  float* buf1 = &wbuf[(wave * 2 + 1) * TKE * WCOLS];

  for (int kc = ks * LEN; kc < ks * LEN + LEN; kc += SUB) {
    // ---- stage hh[0..3][kc .. kc+SUB) = M @ h into LDS ----
    for (int t = tid; t < SUB; t += 256) {
      const int k = kc + t;
      const float h0 = h[0 * DIMV + k];
      const float h1 = h[1 * DIMV + k];
      const float h2 = h[2 * DIMV + k];
      const float h3 = h[3 * DIMV + k];
      hh[0 * PITCH + t] = M00 * h0 + M01 * h1 + M02 * h2 + M02 * h3;
      hh[1 * PITCH + t] = M01 * h0 + 0.5f * h1;
      hh[2 * PITCH + t] = M02 * h0 + Mt  * h2 + Mt  * h3;
      hh[3 * PITCH + t] = M02 * h0 + Mt  * h2 + Mt  * h3;
    }
    __syncthreads();

    const float* Wg = W + (size_t)kc * DIMV + j;       // this chunk's W stripe
    issue_wbuf<TKE>(Wg, buf0, rh, nl);                 // prologue: 2 buffers queued
    issue_wbuf<TKE>(Wg + (size_t)TKE * DIMV, buf1, rh, nl);

    const float* ap = &hh[arow * PITCH + khalf];

#pragma unroll 1
    for (int bi = 0; bi < NB; ++bi) {
      if (bi < NB - 1) wait_async<TKE / 2>();          // cur done, next may fly
      else             wait_async<0>();                // drain for last buffer
      float* cur = (bi & 1) ? buf1 : buf0;
      const float* bp = cur + khalf * WCOLS + 4 * nl;
#pragma unroll
      for (int kk = 0; kk < TKE; kk += 4) {
        v2f a  = *reinterpret_cast<const v2f*>(ap); ap += 4;  // ds_load_b64 (hh)
        v4f w0 = *reinterpret_cast<const v4f*>(bp);           // ds_load_b128 row k+khalf
        v4f w1 = *reinterpret_cast<const v4f*>(bp + WCOLS);   // ds_load_b128 row k+khalf+1
        v2f b0 = {w0.x, w1.x};
        v2f b1 = {w0.y, w1.y};
        v2f b2 = {w0.z, w1.z};
        v2f b3 = {w0.w, w1.w};
        c0 = __builtin_amdgcn_wmma_f32_16x16x4_f32(false, a, false, b0, (short)0, c0, false, false);
        c1 = __builtin_amdgcn_wmma_f32_16x16x4_f32(false, a, false, b1, (short)0, c1, false, false);
        c2 = __builtin_amdgcn_wmma_f32_16x16x4_f32(false, a, false, b2, (short)0, c2, false, false);
        c3 = __builtin_amdgcn_wmma_f32_16x16x4_f32(false, a, false, b3, (short)0, c3, false, false);
        bp += 4 * WCOLS;
      }
      if (bi + 2 < NB)                                  // refill consumed buffer
        issue_wbuf<TKE>(Wg + (size_t)(bi + 2) * TKE * DIMV, cur, rh, nl);
    }
    __syncthreads();
  }

  // D layout: VGPR r, lanes 0-15 = D[M=r][N=lane]; lane n, row r owns
  // 4 consecutive output columns j+4n+{0..3} (one per tile).
  if (lane < 16) {
    const size_t cb = (size_t)j + 4 * nl;
    for (int r = 0; r < 4; ++r) {
      v4f v = {c0[r], c1[r], c2[r], c3[r]};
      *reinterpret_cast<v4f*>(dst + (size_t)(ks * 4 + r) * DIMV + cb) = v;  // b128
    }
  }
}

// Fixed-order sum of the NSPLIT K-slice partials, add bias, tanh.
template <int NSPLIT>
__global__ __launch_bounds__(256) void gcn_epilogue(
    const float* __restrict__ ws,
    const float* __restrict__ bias,
    float* __restrict__ out)
{
  const int idx = blockIdx.x * 256 + threadIdx.x;   // 0..32767
  const int c = idx & (DIMV - 1);
  const int r = idx >> 13;
  float s = 0.0f;
#pragma unroll
  for (int q = 0; q < NSPLIT; ++q)
    s += ws[(size_t)(q * 4 + r) * DIMV + c];
  out[idx] = tanhf(s + bias[c]);
}

extern "C" void kernel_launch(void* const* d_in, const int* in_sizes, int n_in,
                              void* d_out, int out_size, void* d_ws, size_t ws_size,
                              hipStream_t stream) {
  // setup_inputs order: t (unused), h [4*8192], W [8192*8192], b [8192]
  const float* h = (const float*)d_in[1];
  const float* W = (const float*)d_in[2];
  const float* b = (const float*)d_in[3];
  float* out = (float*)d_out;
  float* ws  = (float*)d_ws;

  if (d_ws != nullptr && ws_size >= (size_t)16 * 4 * DIMV * sizeof(float)) {
    // 2048 waves: 128 column-groups x 16 K-slices, deep async HBM pipelining.
    gcn_partial_wmma<16><<<dim3(16 * 16), dim3(256), 0, stream>>>(h, W, ws);
    gcn_epilogue<16><<<dim3(32768 / 256), dim3(256), 0, stream>>>(ws, b, out);
  } else if (d_ws != nullptr && ws_size >= (size_t)4 * 4 * DIMV * sizeof(float)) {
    gcn_partial_wmma<4><<<dim3(4 * 16), dim3(256), 0, stream>>>(h, W, ws);
    gcn_epilogue<4><<<dim3(32768 / 256), dim3(256), 0, stream>>>(ws, b, out);
  } else {
    // No usable workspace: full sums straight into d_out, then in-place bias+tanh.
    gcn_partial_wmma<1><<<dim3(16), dim3(256), 0, stream>>>(h, W, out);
    gcn_epilogue<1><<<dim3(32768 / 256), dim3(256), 0, stream>>>(out, b, out);
  }
}